// GCN_GRU_68925635166716
// MI455X (gfx1250) — compile-verified
//
#include <hip/hip_runtime.h>
#include <hip/hip_bf16.h>
#include <math.h>

// ---------------------------------------------------------------------------
// GCN-GRU (DCRNN-style) for MI455X / gfx1250.
// B=32, T=12, N=512, C=1, H=64, K=3, din=65 (pad->80), horizon=12.
// Heavy op: per gate, K*B matmuls (512x512)@(512x65), fp32.
// V_WMMA_F32_16X16X4_F32 (exact fp32); G and W pre-packed into fragment
// order once per launch; X staged transposed in LDS (conflict-free pitches).
// ---------------------------------------------------------------------------

typedef float v2f __attribute__((ext_vector_type(2)));
typedef float v8f __attribute__((ext_vector_type(8)));

#define NN 512
#define HH 64
#define XPITCH 522   // LDS pitch for Xt[65][XPITCH]   (bank-conflict free)
#define SPITCH 242   // LDS pitch for S [64][SPITCH]   (bank-conflict free)
#define HPITCH 66
#define THREADS 384
#define WAVES 12

__device__ __forceinline__ v8f wmma4(v2f a, v2f b, v8f c) {
  // D = A(16x4 f32) * B(4x16 f32) + C(16x16 f32)
  return __builtin_amdgcn_wmma_f32_16x16x4_f32(false, a, false, b,
                                               (short)0, c, false, false);
}

// ---------------------------------------------------------------------------
// Core GCN matmul: S[64][240] = concat_k( G_k[rows,:] @ X[:,0:80] ).
// Wave w handles (mt = w/3, k = w%3); 5 column tiles of 16; 128 K-steps.
// A fragments pre-packed (coalesced b128); B fragments from LDS Xt
// (transposed: Xt[p][j]).  Tile 4 covers p=64..79; only p==64 is real -> the
// other lanes are zeroed with a multiplicative 0/1 mask (no EXEC divergence).
// ---------------------------------------------------------------------------
__device__ __forceinline__ void gcn_wmma(const float* __restrict__ Gp,
                                         const float* xt, float* sp,
                                         int rb, int w, int lane) {
  const int k = w % 3, mt = w / 3;
  const int lo = lane & 15, hi = lane >> 4;
  const float msk4 = (lo == 0) ? 1.f : 0.f;
  v8f acc[5] = {};
  const float4* gsrc =
      (const float4*)(Gp + (size_t)(k * 32 + rb * 4 + mt) * 8192) + lane;
#pragma unroll 2
  for (int jp = 0; jp < 64; ++jp) {
    float4 a = gsrc[(size_t)jp * 32];           // A frags for 2 K-steps
    v2f a0 = {a.x, a.y}, a1 = {a.z, a.w};
    const int j0 = jp * 8 + hi * 2;
#pragma unroll
    for (int nt = 0; nt < 5; ++nt) {
      v2f b0, b1;
      if (nt < 4) {
        const float* xb = &xt[(nt * 16 + lo) * XPITCH + j0];
        b0 = {xb[0], xb[1]};
        b1 = {xb[4], xb[5]};
      } else {                       // p = 64 + lo; only lo==0 column is real
        const float* xb = &xt[64 * XPITCH + j0];
        b0 = {xb[0] * msk4, xb[1] * msk4};
        b1 = {xb[4] * msk4, xb[5] * msk4};
      }
      acc[nt] = wmma4(a0, b0, acc[nt]);
      acc[nt] = wmma4(a1, b1, acc[nt]);
    }
  }
  __syncthreads();                               // all Xt reads done (union!)
#pragma unroll
  for (int nt = 0; nt < 5; ++nt)
#pragma unroll
    for (int v = 0; v < 8; ++v)
      sp[(mt * 16 + v + 8 * hi) * SPITCH + k * 80 + nt * 16 + lo] = acc[nt][v];
  __syncthreads();
}

// ---------------------------------------------------------------------------
// Pack kernels (run once per launch; weights/graph are loop constants).
// ---------------------------------------------------------------------------
__global__ void k_pack_g(const float* __restrict__ G, float* __restrict__ Gp) {
  int idx = blockIdx.x * 256 + threadIdx.x;     // ((k*32+mt)*64+jp)*128+L*4+q
  if (idx >= 3 * 32 * 64 * 128) return;
  int q = idx & 3, L = (idx >> 2) & 31, jp = (idx >> 7) & 63;
  int mt = (idx >> 13) & 31, k = idx >> 18;
  int i = mt * 16 + (L & 15);
  int j = 8 * jp + 4 * (q >> 1) + 2 * (L >> 4) + (q & 1);
  Gp[idx] = G[((size_t)k * NN + i) * NN + j];
}

__global__ void k_pack_w(const float* __restrict__ W, float* __restrict__ Wpk,
                         int ncols, int total) {
  int idx = blockIdx.x * 256 + threadIdx.x;     // ((nt*30+jp)*32+L)*4+q
  if (idx >= total) return;
  int q = idx & 3, L = (idx >> 2) & 31;
  int jp = (idx >> 7) % 30, nt = (idx >> 7) / 30;
  int col = nt * 16 + (L & 15);
  int rowp = 8 * jp + 4 * (q >> 1) + 2 * (L >> 4) + (q & 1);   // 0..239 padded
  int k = rowp / 80, p = rowp % 80;
  Wpk[idx] = (p < 65) ? W[(size_t)(k * 65 + p) * ncols + col] : 0.f;
}

// ---------------------------------------------------------------------------
// Kernel 1: zr = sigmoid(GCN([x,h]) @ Wg + bg); write z and x2h = r*h.
// ---------------------------------------------------------------------------
struct SmemGate { union { float xt[65 * XPITCH]; float sp[64 * SPITCH]; }; };

__global__ __launch_bounds__(THREADS) void k_gate(
    const float* __restrict__ Gp, const float* __restrict__ Wgp,
    const float* __restrict__ bg, const float* __restrict__ xin,
    int xstride, int xoff, const float* __restrict__ h,
    float* __restrict__ zws, float* __restrict__ x2h) {
  __shared__ SmemGate sm;
  const int b = blockIdx.y, rb = blockIdx.x;
  const int tid = threadIdx.x, w = tid >> 5, lane = tid & 31;
  const int lo = lane & 15, hi = lane >> 4;

  // Stage Xt[p][j]: row 0 = x, rows 1..64 = h (float4, coalesced).
  for (int idx = tid; idx < NN * HH / 4; idx += THREADS) {
    int j = idx >> 4, c4 = (idx & 15) * 4;
    float4 hv = *(const float4*)&h[((size_t)b * NN + j) * HH + c4];
    sm.xt[(c4 + 1) * XPITCH + j] = hv.x;
    sm.xt[(c4 + 2) * XPITCH + j] = hv.y;
    sm.xt[(c4 + 3) * XPITCH + j] = hv.z;
    sm.xt[(c4 + 4) * XPITCH + j] = hv.w;
  }
  for (int j = tid; j < NN; j += THREADS)
    sm.xt[j] = xin[(size_t)b * xstride + xoff + j];
  __syncthreads();

  gcn_wmma(Gp, sm.xt, sm.sp, rb, w, lane);

  // Gate matmul: (64x240) @ (240x128); 32 tiles over 12 waves.
  for (int t = w; t < 32; t += WAVES) {
    int omt = t >> 3, ont = t & 7;
    v8f acc2 = {};
    const float4* wsrc = (const float4*)(Wgp + (size_t)ont * 30 * 128) + lane;
    const float* arow = &sm.sp[(omt * 16 + lo) * SPITCH + 2 * hi];
    for (int jp = 0; jp < 30; ++jp) {
      float4 wb = wsrc[(size_t)jp * 32];
      v2f b0 = {wb.x, wb.y}, b1 = {wb.z, wb.w};
      const float* ap = arow + jp * 8;
      v2f a0 = {ap[0], ap[1]}, a1 = {ap[4], ap[5]};
      acc2 = wmma4(a0, b0, acc2);
      acc2 = wmma4(a1, b1, acc2);
    }
    const int col = ont * 16 + lo;
    const float bgv = bg[col];
    const int r0 = rb * 64 + omt * 16;
#pragma unroll
    for (int v = 0; v < 8; ++v) {
      int row = r0 + v + 8 * hi;
      float g = 1.f / (1.f + __expf(-(acc2[v] + bgv)));
      if (ont < 4) {                                   // z half
        zws[((size_t)b * NN + row) * HH + col] = g;
      } else {                                         // r half -> x2h = r*h
        int cc = col - 64;
        float hv = h[((size_t)b * NN + row) * HH + cc];
        x2h[((size_t)b * NN + row) * HH + cc] = g * hv;
      }
    }
  }
}

// ---------------------------------------------------------------------------
// Kernel 2: hc = tanh(GCN([x, x2h]) @ Wu + bu); h = (1-z)h + z*hc;
// decoder: go = h @ Wp + bp.
// ---------------------------------------------------------------------------
struct SmemUpd {
  union { float xt[65 * XPITCH]; float sp[64 * SPITCH]; } u;
  float hst[64 * HPITCH];
};

__global__ __launch_bounds__(THREADS) void k_update(
    const float* __restrict__ Gp, const float* __restrict__ Wup,
    const float* __restrict__ bu, const float* __restrict__ xin,
    int xstride, int xoff, const float* __restrict__ x2h,
    float* __restrict__ h, const float* __restrict__ zws,
    const float* __restrict__ Wp, const float* __restrict__ bp,
    float* __restrict__ gows, float* __restrict__ out, int dec) {
  __shared__ SmemUpd sm;
  const int b = blockIdx.y, rb = blockIdx.x;
  const int tid = threadIdx.x, w = tid >> 5, lane = tid & 31;
  const int lo = lane & 15, hi = lane >> 4;

  for (int idx = tid; idx < NN * HH / 4; idx += THREADS) {
    int j = idx >> 4, c4 = (idx & 15) * 4;
    float4 hv = *(const float4*)&x2h[((size_t)b * NN + j) * HH + c4];
    sm.u.xt[(c4 + 1) * XPITCH + j] = hv.x;
    sm.u.xt[(c4 + 2) * XPITCH + j] = hv.y;
    sm.u.xt[(c4 + 3) * XPITCH + j] = hv.z;
    sm.u.xt[(c4 + 4) * XPITCH + j] = hv.w;
  }
  for (int j = tid; j < NN; j += THREADS)
    sm.u.xt[j] = xin[(size_t)b * xstride + xoff + j];
  __syncthreads();

  gcn_wmma(Gp, sm.u.xt, sm.u.sp, rb, w, lane);

  // Update matmul: (64x240) @ (240x64); 16 tiles over 12 waves.
  for (int t = w; t < 16; t += WAVES) {
    int omt = t >> 2, ont = t & 3;
    v8f acc2 = {};
    const float4* wsrc = (const float4*)(Wup + (size_t)ont * 30 * 128) + lane;
    const float* arow = &sm.u.sp[(omt * 16 + lo) * SPITCH + 2 * hi];
    for (int jp = 0; jp < 30; ++jp) {
      float4 wb = wsrc[(size_t)jp * 32];
      v2f b0 = {wb.x, wb.y}, b1 = {wb.z, wb.w};
      const float* ap = arow + jp * 8;
      v2f a0 = {ap[0], ap[1]}, a1 = {ap[4], ap[5]};
      acc2 = wmma4(a0, b0, acc2);
      acc2 = wmma4(a1, b1, acc2);
    }
    const int col = ont * 16 + lo;
    const float buv = bu[col];
#pragma unroll
    for (int v = 0; v < 8; ++v) {
      int rl = omt * 16 + v + 8 * hi;
      int row = rb * 64 + rl;
      size_t ix = ((size_t)b * NN + row) * HH + col;
      float hc = tanhf(acc2[v] + buv);
      float z = zws[ix];
      float hv = h[ix];
      float hn = hv + z * (hc - hv);                   // (1-z)h + z*hc
      h[ix] = hn;
      sm.hst[rl * HPITCH + col] = hn;
    }
  }
  __syncthreads();
  if (dec >= 0 && tid < 64) {                          // decoder projection
    int row = rb * 64 + tid;
    float s = bp[0];
    for (int c2 = 0; c2 < HH; ++c2) s += sm.hst[tid * HPITCH + c2] * Wp[c2];
    gows[(size_t)b * NN + row] = s;
    out[((size_t)b * 12 + dec) * NN + row] = s;
  }
}

// ---------------------------------------------------------------------------
extern "C" void kernel_launch(void* const* d_in, const int* in_sizes, int n_in,
                              void* d_out, int out_size, void* d_ws,
                              size_t ws_size, hipStream_t stream) {
  (void)in_sizes; (void)n_in; (void)out_size; (void)ws_size;
  const float* x    = (const float*)d_in[0];
  const float* P    = (const float*)d_in[1];
  const float* We_g = (const float*)d_in[2];
  const float* be_g = (const float*)d_in[3];
  const float* We_u = (const float*)d_in[4];
  const float* be_u = (const float*)d_in[5];
  const float* Wd_g = (const float*)d_in[6];
  const float* bd_g = (const float*)d_in[7];
  const float* Wd_u = (const float*)d_in[8];
  const float* bd_u = (const float*)d_in[9];
  const float* Wp   = (const float*)d_in[10];
  const float* bp   = (const float*)d_in[11];
  float* out = (float*)d_out;

  float* ws   = (float*)d_ws;
  float* Gp   = ws;                  // 786432
  float* Wge  = Gp  + 786432;        // 30720
  float* Wue  = Wge + 30720;         // 15360
  float* Wgd  = Wue + 15360;         // 30720
  float* Wud  = Wgd + 30720;         // 15360
  float* hws  = Wud + 15360;         // 1048576  (B,N,H)
  float* zws  = hws + 1048576;       // 1048576  (B,N,H)
  float* x2h  = zws + 1048576;       // 1048576  (B,N,H)  r*h
  float* gows = x2h + 1048576;       // 16384    (B,N)

  hipMemsetAsync(hws, 0, 1048576 * sizeof(float), stream);
  hipMemsetAsync(gows, 0, 16384 * sizeof(float), stream);

  k_pack_g<<<(786432 + 255) / 256, 256, 0, stream>>>(P, Gp);
  k_pack_w<<<(30720 + 255) / 256, 256, 0, stream>>>(We_g, Wge, 128, 30720);
  k_pack_w<<<(15360 + 255) / 256, 256, 0, stream>>>(We_u, Wue, 64, 15360);
  k_pack_w<<<(30720 + 255) / 256, 256, 0, stream>>>(Wd_g, Wgd, 128, 30720);
  k_pack_w<<<(15360 + 255) / 256, 256, 0, stream>>>(Wd_u, Wud, 64, 15360);

  dim3 grid(8, 32);
  for (int t = 0; t < 12; ++t) {                       // encoder
    k_gate<<<grid, THREADS, 0, stream>>>(Gp, Wge, be_g, x, 12 * NN, t * NN,
                                         hws, zws, x2h);
    k_update<<<grid, THREADS, 0, stream>>>(Gp, Wue, be_u, x, 12 * NN, t * NN,
                                           x2h, hws, zws, Wp, bp, gows, out, -1);
  }
  for (int d = 0; d < 12; ++d) {                       // decoder
    k_gate<<<grid, THREADS, 0, stream>>>(Gp, Wgd, bd_g, gows, NN, 0,
                                         hws, zws, x2h);
    k_update<<<grid, THREADS, 0, stream>>>(Gp, Wud, bd_u, gows, NN, 0,
                                           x2h, hws, zws, Wp, bp, gows, out, d);
  }
}